// MultiHeadAttentionQuantumHybrid_65481071409607
// MI455X (gfx1250) — compile-verified
//
#include <hip/hip_runtime.h>
#include <math.h>

// ---------------------------------------------------------------------------
// MultiHeadAttentionQuantumHybrid for MI455X (gfx1250, wave32, WMMA)
//
// Fused algebra:
//   Wc[j][h*8+qi, e]   = sum_d Pin_j[qi,d] * Wj[h*64+d, e]        (192 x 512)
//   P  = x @ Wc_all^T                                             (1024 x 192)  [WMMA f32]
//   Z  = quantum_circuit(P) rowwise per 8 angles                  (1024 x 192)  [wave32/shfl]
//   G  = Pout_q^T @ Pout_k / sqrt(dk)                              (8 x 8)
//   scores = z_q G z_k^T ; softmax ; U = attn @ z_v                (per b,h)     [wave/row]
//   WfuseT[e, h*8+qi]  = sum_d Pout_v[d,qi] * Wo[e, h*64+d]        (512 x 64)
//   out = U @ WfuseT^T                                            (1024 x 512)  [WMMA f32]
// ---------------------------------------------------------------------------

typedef __attribute__((ext_vector_type(2))) float v2f;
typedef __attribute__((ext_vector_type(8))) float v8f;

#define E_DIM 512
#define S_DIM 512
#define B_DIM 2
#define H_DIM 8
#define DK    64
#define NQ    8
#define MROWS (B_DIM * S_DIM)       // 1024
#define NCOLS (3 * H_DIM * NQ)      // 192

// ---------------------------------------------------------------------------
// Kernel 0: precompute fused weight matrices, Gram matrix, theta trig.
// ---------------------------------------------------------------------------
__global__ void qmha_precompute(const float* __restrict__ Wq, const float* __restrict__ Wk,
                                const float* __restrict__ Wv, const float* __restrict__ Wo,
                                const float* __restrict__ Pin_q, const float* __restrict__ Pin_k,
                                const float* __restrict__ Pin_v, const float* __restrict__ Pout_q,
                                const float* __restrict__ Pout_k, const float* __restrict__ Pout_v,
                                const float* __restrict__ theta,
                                float* __restrict__ wc, float* __restrict__ wft,
                                float* __restrict__ g, float* __restrict__ trc,
                                float* __restrict__ trs) {
  const int WC_N = NCOLS * E_DIM;   // 98304
  const int WFT_N = E_DIM * 64;     // 32768
  int idx = blockIdx.x * blockDim.x + threadIdx.x;
  if (idx < WC_N) {
    int n = idx >> 9;               // row (0..191)
    int e = idx & 511;
    int j = n >> 6;                 // 0=q 1=k 2=v
    int h = (n >> 3) & 7;
    int qi = n & 7;
    const float* W = (j == 0) ? Wq : ((j == 1) ? Wk : Wv);
    const float* Pin = (j == 0) ? Pin_q : ((j == 1) ? Pin_k : Pin_v);
    float acc = 0.f;
    for (int d = 0; d < DK; ++d)
      acc += Pin[qi * DK + d] * W[(h * DK + d) * E_DIM + e];
    wc[n * E_DIM + e] = acc;
  } else if (idx < WC_N + WFT_N) {
    int t = idx - WC_N;
    int e = t >> 6;
    int n = t & 63;
    int h = n >> 3;
    int qi = n & 7;
    float acc = 0.f;
    for (int d = 0; d < DK; ++d)
      acc += Pout_v[d * NQ + qi] * Wo[e * E_DIM + h * DK + d];
    wft[e * 64 + n] = acc;
  } else if (idx < WC_N + WFT_N + 64) {
    int t = idx - WC_N - WFT_N;
    int qi = t >> 3;
    int qj = t & 7;
    float acc = 0.f;
    for (int d = 0; d < DK; ++d)
      acc += Pout_q[d * NQ + qi] * Pout_k[d * NQ + qj];
    g[qi * 8 + qj] = acc * 0.125f;  // 1/sqrt(dk) = 1/8 folded in
  } else if (idx < WC_N + WFT_N + 64 + 16) {
    int t = idx - WC_N - WFT_N - 64;
    float th = theta[t] * 0.5f;
    trc[t] = cosf(th);
    trs[t] = sinf(th);
  }
}

// ---------------------------------------------------------------------------
// WMMA f32 GEMM:  C[M,N] = A[M,K] @ Bm[N,K]^T    (one wave per 16x16 tile)
// Grid must supply exactly (M/16)*(N/16) waves.
// Fragment layout per CDNA5 ISA 7.12.2:
//   A 16x4 f32: lanes 0-15 hold (m=lane, K=0..1), lanes 16-31 hold K=2..3
//   B 4x16 f32: VGPR r: lanes 0-15 -> row K=r, lanes 16-31 -> row K=r+2
//   C 16x16  : VGPR r: lanes 0-15 -> M=r, lanes 16-31 -> M=r+8, N=lane&15
// ---------------------------------------------------------------------------
__global__ void qmha_gemm_xt_wmma(const float* __restrict__ A, const float* __restrict__ Bm,
                                  float* __restrict__ C, int N, int K) {
  int wave = blockIdx.x * (blockDim.x >> 5) + (threadIdx.x >> 5);
  int lane = threadIdx.x & 31;
  int tilesN = N >> 4;
  int tm = (wave / tilesN) << 4;
  int tn = (wave % tilesN) << 4;
  int half = lane >> 4;       // 0: K 0..1, 1: K 2..3
  int l15 = lane & 15;

  const float* arow = A + (size_t)(tm + l15) * K + half * 2;
  const float* brow = Bm + (size_t)(tn + l15) * K + half * 2;

  v8f acc = {};
  for (int k = 0; k < K; k += 4) {
    v2f a = *(const v2f*)(arow + k);
    v2f b = *(const v2f*)(brow + k);
    // 8 args: (neg_a, A, neg_b, B, c_mod, C, reuse_a, reuse_b)
    acc = __builtin_amdgcn_wmma_f32_16x16x4_f32(false, a, false, b, (short)0, acc,
                                                false, false);
  }
  float* cbase = C + (size_t)(tm + half * 8) * N + tn + l15;
#pragma unroll
  for (int r = 0; r < 8; ++r)
    cbase[(size_t)r * N] = acc[r];
}

// ---------------------------------------------------------------------------
// Quantum circuit kernel: one wave32 per row of 8 angles.
// State: 256 complex amplitudes, 8 per lane; index i = (j<<5)|lane,
// wire w <-> bit (7-w) (wire 0 = MSB, matching the reference reshape).
// Initial product state amplitude = mag * (-i)^popcount(i).
// RY (real 2x2) on wire w: bit 7-w. bits 5..7 = register-local, 0..4 = shfl_xor.
// CNOT chain folded into the final expectation signs (parity of leading bits).
// ---------------------------------------------------------------------------
__global__ void qmha_circuit(const float* __restrict__ P, const float* __restrict__ trc,
                             const float* __restrict__ trs, float* __restrict__ Z) {
  int wave = blockIdx.x * (blockDim.x >> 5) + (threadIdx.x >> 5);
  int lane = threadIdx.x & 31;
  int m = wave / 24;            // row in [0, 1024)
  int grp = wave % 24;          // group of 8 angles (j*8 + h)

  const float* ang = P + (size_t)m * NCOLS + grp * 8;
  float c[8], s[8];
#pragma unroll
  for (int w = 0; w < 8; ++w) {
    float a = ang[w] * 0.5f;
    c[w] = __cosf(a);
    s[w] = __sinf(a);
  }

  float re[8], im[8];
#pragma unroll
  for (int j = 0; j < 8; ++j) {
    int i = (j << 5) | lane;
    float mag = 1.f;
#pragma unroll
    for (int w = 0; w < 8; ++w)
      mag *= ((i >> (7 - w)) & 1) ? s[w] : c[w];
    int p = __popc(i) & 3;      // (-i)^p
    re[j] = (p == 0) ? mag : ((p == 2) ? -mag : 0.f);
    im[j] = (p == 1) ? -mag : ((p == 3) ? mag : 0.f);
  }

  // 2 layers x 8 wires of RY(theta). Fully unrolled so register indices and
  // the local/cross-lane split are compile-time constants.
#pragma unroll
  for (int t = 0; t < 16; ++t) {
    float cc = trc[t];
    float ss = trs[t];
    const int w = t & 7;
    const int p = 7 - w;        // bit position
    if (p >= 5) {               // register-local pairing
      const int str = 1 << (p - 5);
#pragma unroll
      for (int j = 0; j < 8; ++j) {
        if ((j & str) == 0) {
          const int j1 = j | str;
          float r0 = re[j], r1 = re[j1];
          re[j] = cc * r0 - ss * r1;
          re[j1] = ss * r0 + cc * r1;
          float i0 = im[j], i1 = im[j1];
          im[j] = cc * i0 - ss * i1;
          im[j1] = ss * i0 + cc * i1;
        }
      }
    } else {                    // cross-lane pairing via shfl_xor
      const int mask = 1 << p;
      const int mybit = (lane >> p) & 1;
#pragma unroll
      for (int j = 0; j < 8; ++j) {
        float orr = __shfl_xor(re[j], mask, 32);
        float oii = __shfl_xor(im[j], mask, 32);
        re[j] = mybit ? (ss * orr + cc * re[j]) : (cc * re[j] - ss * orr);
        im[j] = mybit ? (ss * oii + cc * im[j]) : (cc * im[j] - ss * oii);
      }
    }
  }

  // probabilities + PauliZ expectations with CNOT-chain parity signs
  float pr[8];
#pragma unroll
  for (int j = 0; j < 8; ++j)
    pr[j] = re[j] * re[j] + im[j] * im[j];

  float ez[8];
#pragma unroll
  for (int w = 0; w < 8; ++w) {
    float acc = 0.f;
#pragma unroll
    for (int j = 0; j < 8; ++j) {
      int i = (j << 5) | lane;
      acc += (__popc(i >> (7 - w)) & 1) ? -pr[j] : pr[j];
    }
    ez[w] = acc;
  }
#pragma unroll
  for (int off = 16; off > 0; off >>= 1) {
#pragma unroll
    for (int w = 0; w < 8; ++w)
      ez[w] += __shfl_xor(ez[w], off, 32);
  }
  if (lane == 0) {
    float4* o = (float4*)(Z + (size_t)m * NCOLS + grp * 8);
    o[0] = make_float4(ez[0], ez[1], ez[2], ez[3]);
    o[1] = make_float4(ez[4], ez[5], ez[6], ez[7]);
  }
}

// ---------------------------------------------------------------------------
// Attention kernel: one wave32 per (b,h,s) query row.
// scores[t] = (z_q . G) . z_k[t]; stable softmax; U = attn @ z_v  (8-wide).
// ---------------------------------------------------------------------------
__global__ void qmha_attention(const float* __restrict__ Z, const float* __restrict__ G,
                               float* __restrict__ U) {
  int wave = blockIdx.x * (blockDim.x >> 5) + (threadIdx.x >> 5);
  int lane = threadIdx.x & 31;
  int s = wave & (S_DIM - 1);
  int h = (wave >> 9) & 7;
  int b = wave >> 12;
  int mq = b * S_DIM + s;

  float y[8];
  {
    const float* zr = Z + (size_t)mq * NCOLS + h * 8;   // j=0 block (q)
    float zq[8];
#pragma unroll
    for (int w = 0; w < 8; ++w) zq[w] = zr[w];
#pragma unroll
    for (int qi = 0; qi < 8; ++qi) {
      float acc = 0.f;
#pragma unroll
      for (int w = 0; w < 8; ++w) acc += zq[w] * G[w * 8 + qi];
      y[qi] = acc;
    }
  }

  const float* Zk = Z + (size_t)(b * S_DIM) * NCOLS + 64 + h * 8;   // j=1 block (k)
  const float* Zv = Z + (size_t)(b * S_DIM) * NCOLS + 128 + h * 8;  // j=2 block (v)

  float sc[16];
  float mx = -3.0e38f;
#pragma unroll 4
  for (int it = 0; it < 16; ++it) {
    int t = it * 32 + lane;
    const float4* p = (const float4*)(Zk + (size_t)t * NCOLS);
    float4 z0 = p[0], z1 = p[1];
    float d = y[0] * z0.x + y[1] * z0.y + y[2] * z0.z + y[3] * z0.w +
              y[4] * z1.x + y[5] * z1.y + y[6] * z1.z + y[7] * z1.w;
    sc[it] = d;
    mx = fmaxf(mx, d);
  }
#pragma unroll
  for (int off = 16; off > 0; off >>= 1)
    mx = fmaxf(mx, __shfl_xor(mx, off, 32));

  float sum = 0.f;
  float u[8] = {0.f, 0.f, 0.f, 0.f, 0.f, 0.f, 0.f, 0.f};
#pragma unroll 4
  for (int it = 0; it < 16; ++it) {
    int t = it * 32 + lane;
    float e = __expf(sc[it] - mx);
    sum += e;
    const float4* p = (const float4*)(Zv + (size_t)t * NCOLS);
    float4 z0 = p[0], z1 = p[1];
    u[0] += e * z0.x; u[1] += e * z0.y; u[2] += e * z0.z; u[3] += e * z0.w;
    u[4] += e * z1.x; u[5] += e * z1.y; u[6] += e * z1.z; u[7] += e * z1.w;
  }
#pragma unroll
  for (int off = 16; off > 0; off >>= 1) {
    sum += __shfl_xor(sum, off, 32);
#pragma unroll
    for (int w = 0; w < 8; ++w) u[w] += __shfl_xor(u[w], off, 32);
  }
  if (lane == 0) {
    float inv = 1.0f / sum;
    float4* o = (float4*)(U + (size_t)mq * 64 + h * 8);
    o[0] = make_float4(u[0] * inv, u[1] * inv, u[2] * inv, u[3] * inv);
    o[1] = make_float4(u[4] * inv, u[5] * inv, u[6] * inv, u[7] * inv);
  }
}

// ---------------------------------------------------------------------------
// Host-side launch. Workspace layout (floats):
//   wc   @ 0        (192*512 = 98304)
//   wft  @ 98304    (512*64  = 32768)
//   G    @ 131072   (64)
//   trc  @ 131136   (16)
//   trs  @ 131152   (16)
//   P    @ 131200   (1024*192 = 196608)
//   Z    @ 327808   (196608)
//   U    @ 524416   (1024*64 = 65536)     total ~2.36 MB
// ---------------------------------------------------------------------------
extern "C" void kernel_launch(void* const* d_in, const int* in_sizes, int n_in,
                              void* d_out, int out_size, void* d_ws, size_t ws_size,
                              hipStream_t stream) {
  const float* x      = (const float*)d_in[0];
  const float* Wq     = (const float*)d_in[1];
  const float* Wk     = (const float*)d_in[2];
  const float* Wv     = (const float*)d_in[3];
  const float* Wo     = (const float*)d_in[4];
  const float* Pin_q  = (const float*)d_in[5];
  const float* Pin_k  = (const float*)d_in[6];
  const float* Pin_v  = (const float*)d_in[7];
  const float* Pout_q = (const float*)d_in[8];
  const float* Pout_k = (const float*)d_in[9];
  const float* Pout_v = (const float*)d_in[10];
  const float* theta  = (const float*)d_in[11];

  float* ws  = (float*)d_ws;
  float* wc  = ws;
  float* wft = ws + 98304;
  float* g   = ws + 131072;
  float* trc = ws + 131136;
  float* trs = ws + 131152;
  float* P   = ws + 131200;
  float* Z   = ws + 327808;
  float* U   = ws + 524416;

  // 0) fused weights (131152 work items)
  qmha_precompute<<<513, 256, 0, stream>>>(Wq, Wk, Wv, Wo, Pin_q, Pin_k, Pin_v,
                                           Pout_q, Pout_k, Pout_v, theta,
                                           wc, wft, g, trc, trs);
  // 1) P = x @ wc^T : (1024 x 192), K=512 -> 768 tiles -> 96 blocks of 8 waves
  qmha_gemm_xt_wmma<<<96, 256, 0, stream>>>(x, wc, P, NCOLS, E_DIM);
  // 2) quantum circuits: 1024*24 = 24576 waves -> 3072 blocks of 8 waves
  qmha_circuit<<<3072, 256, 0, stream>>>(P, trc, trs, Z);
  // 3) attention: 8192 waves -> 1024 blocks
  qmha_attention<<<1024, 256, 0, stream>>>(Z, g, U);
  // 4) out = U @ wft^T : (1024 x 512), K=64 -> 2048 tiles -> 256 blocks
  qmha_gemm_xt_wmma<<<256, 256, 0, stream>>>(U, wft, (float*)d_out, E_DIM, 64);
}